// Classifier_52175262712415
// MI455X (gfx1250) — compile-verified
//
#include <hip/hip_runtime.h>

// ---------------------------------------------------------------------------
// Problem constants (from the reference): N=65536, D=768, L=100, B=8192
// ---------------------------------------------------------------------------

typedef float v2f __attribute__((ext_vector_type(2)));
typedef float v8f __attribute__((ext_vector_type(8)));

// ---------------------------------------------------------------------------
// C[M,N] = A[M,K] @ B[K,N] + bias[N]  via V_WMMA_F32_16X16X4_F32
//
// Block tile: 128 (M) x 128 (N); 8 waves arranged 4 (M) x 2 (N):
//   wave&3  -> 32-row M slab (2 sub-tiles of 16)
//   wave>>2 -> 64-col N half (4 tiles of 16)
// => 8 accumulators (v8f) per wave; each B fragment reused across 2 M tiles.
//
// LDS holds A and B K-chunks (K=16) in *fragment-ready pair layout*:
//   pair p = K-pair {2p, 2p+1}; lane fragment = one aligned float2
//   -> single ds_load_b64 per WMMA operand, no repack movs.
//
// Requires: M % 128 == 0, K % 16 == 0. N guarded (for the N=100 classifier).
// ---------------------------------------------------------------------------
__global__ __launch_bounds__(256) void wmma_gemm_bias(
    const float* __restrict__ A, const float* __restrict__ Bm,
    const float* __restrict__ bias, float* __restrict__ C,
    int M, int N, int K, int lda, int ldb, int ldc)
{
    // pair-major: [pair (8)][row/col (128)] of float2
    __shared__ float lds_a[8 * 128 * 2];   // 8 KB
    __shared__ float lds_b[8 * 128 * 2];   // 8 KB

    const int tid     = threadIdx.x;
    const int lane    = tid & 31;
    const int wave    = tid >> 5;
    const int lmod    = lane & 15;     // row/col within 16x16 tile
    const int halfSel = lane >> 4;     // lane half selects K-pair / M-half
    const int mslab   = (wave & 3) * 32;
    const int nhalf   = (wave >> 2) * 64;
    const int rowBase = blockIdx.y * 128;
    const int colBase = blockIdx.x * 128;

    v8f acc[2][4] = {};                // [M sub-tile][N tile]

    for (int k0 = 0; k0 < K; k0 += 16) {
        // --- stage A chunk: 128 rows x 16 K -> pair layout ---
        #pragma unroll
        for (int k = 0; k < 8; ++k) {
            int e = tid + k * 256;               // 2048 elements
            int r = e >> 4, kc = e & 15;         // row, k-within-chunk
            lds_a[(((kc >> 1) * 128 + r) << 1) | (kc & 1)] =
                A[(size_t)(rowBase + r) * lda + k0 + kc];
        }
        // --- stage B chunk: 16 K x 128 cols -> pair layout, col-guarded ---
        #pragma unroll
        for (int k = 0; k < 8; ++k) {
            int e = tid + k * 256;               // 2048 elements
            int kc = e >> 7, c = e & 127;        // k-within-chunk, col
            int gc = colBase + c;
            lds_b[(((kc >> 1) * 128 + c) << 1) | (kc & 1)] =
                (gc < N) ? Bm[(size_t)(k0 + kc) * ldb + gc] : 0.0f;
        }
        __syncthreads();

        const v2f* pa = (const v2f*)lds_a;
        const v2f* pb = (const v2f*)lds_b;

        // --- 4 wmma steps of K=4; lane half takes pair (kk/2 + halfSel) ---
        #pragma unroll
        for (int kk = 0; kk < 16; kk += 4) {
            const int p = (kk >> 1) + halfSel;   // pair index for this lane
            v2f af0 = pa[p * 128 + mslab + lmod];
            v2f af1 = pa[p * 128 + mslab + 16 + lmod];
            #pragma unroll
            for (int t = 0; t < 4; ++t) {
                v2f bf = pb[p * 128 + nhalf + t * 16 + lmod];
                acc[0][t] = __builtin_amdgcn_wmma_f32_16x16x4_f32(
                    false, af0, false, bf, (short)0, acc[0][t], false, false);
                acc[1][t] = __builtin_amdgcn_wmma_f32_16x16x4_f32(
                    false, af1, false, bf, (short)0, acc[1][t], false, false);
            }
        }
        __syncthreads();
    }

    // --- bias + store: VGPR r holds M = r + 8*halfSel, col = lmod (ISA 7.12.2) ---
    #pragma unroll
    for (int t = 0; t < 4; ++t) {
        int col = colBase + nhalf + t * 16 + lmod;
        if (col < N) {
            float bv = bias[col];
            #pragma unroll
            for (int m = 0; m < 2; ++m) {
                #pragma unroll
                for (int r = 0; r < 8; ++r) {
                    int row = rowBase + mslab + m * 16 + r + halfSel * 8;
                    C[(size_t)row * ldc + col] = acc[m][t][r] + bv;
                }
            }
        }
    }
}

// ---------------------------------------------------------------------------
// Kernel 2: logit[n] = dot(att_weight[query[n]], pooled[n]); one wave per row
// ---------------------------------------------------------------------------
__global__ __launch_bounds__(256) void logit_kernel(
    const float* __restrict__ pooled, const float* __restrict__ att,
    const int* __restrict__ query, float* __restrict__ logit, int Nrows, int D)
{
    int wave = threadIdx.x >> 5, lane = threadIdx.x & 31;
    int row = blockIdx.x * 8 + wave;
    if (row >= Nrows) return;
    const float* pr = pooled + (size_t)row * D;
    const float* ar = att + (size_t)query[row] * D;
    float acc = 0.0f;
    for (int d = lane; d < D; d += 32) acc += pr[d] * ar[d];
    #pragma unroll
    for (int off = 16; off > 0; off >>= 1) acc += __shfl_down(acc, off, 32);
    if (lane == 0) logit[row] = acc;
}

// ---------------------------------------------------------------------------
// Kernel 3: bag boundaries from sorted seg_ids -> starts[B+1]
// ---------------------------------------------------------------------------
__global__ void seg_starts(const int* __restrict__ seg, int* __restrict__ starts,
                           int Nrows, int B)
{
    int n = blockIdx.x * blockDim.x + threadIdx.x;
    if (n >= Nrows) return;
    int s  = seg[n];
    int sp = (n == 0) ? -1 : seg[n - 1];
    for (int b = sp + 1; b <= s; ++b) starts[b] = n;
    if (n == Nrows - 1)
        for (int b = s + 1; b <= B; ++b) starts[b] = Nrows;
}

// ---------------------------------------------------------------------------
// Kernel 4: per-bag softmax over logits + weighted sum of pooled rows.
// One 256-thread block per bag; D must be 768 (3 dims per thread).
// ---------------------------------------------------------------------------
__global__ __launch_bounds__(256) void bag_reduce(
    const float* __restrict__ pooled, const float* __restrict__ logit,
    const int* __restrict__ starts, float* __restrict__ bag_repre, int D)
{
    __shared__ float red[256];
    __shared__ float wc[256];
    __shared__ float mz[2];
    const int b = blockIdx.x;
    const int tid = threadIdx.x;
    const int s = starts[b], e = starts[b + 1];

    if (s >= e) {   // empty bag -> zero representation
        for (int d = tid; d < D; d += 256) bag_repre[(size_t)b * D + d] = 0.0f;
        return;
    }
    // bag max
    float lm = -3.4e38f;
    for (int i = s + tid; i < e; i += 256) lm = fmaxf(lm, logit[i]);
    red[tid] = lm; __syncthreads();
    for (int o = 128; o > 0; o >>= 1) {
        if (tid < o) red[tid] = fmaxf(red[tid], red[tid + o]);
        __syncthreads();
    }
    if (tid == 0) mz[0] = red[0];
    __syncthreads();
    const float m = mz[0];
    // sum of exp
    float ls = 0.0f;
    for (int i = s + tid; i < e; i += 256) ls += __expf(logit[i] - m);
    red[tid] = ls; __syncthreads();
    for (int o = 128; o > 0; o >>= 1) {
        if (tid < o) red[tid] += red[tid + o];
        __syncthreads();
    }
    if (tid == 0) mz[1] = red[0];
    __syncthreads();
    const float z = mz[1];

    // weighted sum of pooled rows; weights cached in LDS per 256-row chunk
    float acc0 = 0.0f, acc1 = 0.0f, acc2 = 0.0f;
    for (int base = s; base < e; base += 256) {
        int idx = base + tid;
        wc[tid] = (idx < e) ? __expf(logit[idx] - m) / z : 0.0f;
        __syncthreads();
        int cnt = e - base; if (cnt > 256) cnt = 256;
        for (int j = 0; j < cnt; ++j) {
            const float* pr = pooled + (size_t)(base + j) * D;
            float wj = wc[j];
            acc0 += pr[tid]       * wj;
            acc1 += pr[tid + 256] * wj;
            acc2 += pr[tid + 512] * wj;
        }
        __syncthreads();
    }
    bag_repre[(size_t)b * D + tid]       = acc0;
    bag_repre[(size_t)b * D + tid + 256] = acc1;
    bag_repre[(size_t)b * D + tid + 512] = acc2;
}

// ---------------------------------------------------------------------------
// Host-side launcher
// ---------------------------------------------------------------------------
extern "C" void kernel_launch(void* const* d_in, const int* in_sizes, int n_in,
                              void* d_out, int out_size, void* d_ws, size_t ws_size,
                              hipStream_t stream)
{
    const float* h_cls = (const float*)d_in[0];
    const float* W_fc  = (const float*)d_in[1];
    const float* b_fc  = (const float*)d_in[2];
    const float* att   = (const float*)d_in[3];
    const float* W_cls = (const float*)d_in[4];
    const float* b_cls = (const float*)d_in[5];
    const int*   query = (const int*)d_in[6];
    const int*   seg   = (const int*)d_in[7];

    const int N = 65536, D = 768, L = 100, B = 8192;
    (void)in_sizes; (void)n_in; (void)out_size; (void)ws_size;

    // workspace layout (~226 MB)
    char* ws = (char*)d_ws;
    size_t off = 0;
    float* pooled = (float*)(ws + off); off += (size_t)N * D * sizeof(float);
    float* logit  = (float*)(ws + off); off += (size_t)N * sizeof(float);
    int*   starts = (int*)  (ws + off); off += (size_t)(B + 1) * sizeof(int);
    off = (off + 255) & ~(size_t)255;
    float* bagr   = (float*)(ws + off); off += (size_t)B * D * sizeof(float);

    // 1) pooled = h_cls @ W_fc + b_fc      [65536 x 768]
    wmma_gemm_bias<<<dim3(D / 128, N / 128), 256, 0, stream>>>(
        h_cls, W_fc, b_fc, pooled, N, D, D, D, D, D);

    // 2) attention logits                   [65536]
    logit_kernel<<<N / 8, 256, 0, stream>>>(pooled, att, query, logit, N, D);

    // 3) bag boundaries from sorted seg_ids
    seg_starts<<<(N + 255) / 256, 256, 0, stream>>>(seg, starts, N, B);

    // 4) segmented softmax + weighted bag representation  [8192 x 768]
    bag_reduce<<<B, 256, 0, stream>>>(pooled, logit, starts, bagr, D);

    // 5) logits = bag_repre @ W_cls + b_cls  [8192 x 100]
    wmma_gemm_bias<<<dim3((L + 127) / 128, B / 128), 256, 0, stream>>>(
        bagr, W_cls, b_cls, (float*)d_out, B, L, D, D, L, L);
}